// GCN_NET_56891136803140
// MI455X (gfx1250) — compile-verified
//
#include <hip/hip_runtime.h>
#include <math.h>

// ---------------------------------------------------------------------------
// GCN 2-layer forward for MI455X (gfx1250).
//   layer: H = X*W  ->  Agg[d] += H[s] * dinv[s]*dinv[d]  (+ self loops) -> +b
//   GEMMs via V_WMMA_F32_16X16X4_F32 (fp32 WMMA, full precision; GEMM cost
//   is negligible vs. the ~1.5 GB of edge traffic which bounds this problem).
// ---------------------------------------------------------------------------

typedef __attribute__((ext_vector_type(2))) float v2f;
typedef __attribute__((ext_vector_type(8))) float v8f;

static inline int ceil_div(long long a, long long b) { return (int)((a + b - 1) / b); }

// One 16x16 C tile per wave32. K stepped by 4 via V_WMMA_F32_16X16X4_F32.
// A: [M,K] row-major   B: [K,N] row-major   C: [M,N] row-major
template <int K, int N>
__global__ __launch_bounds__(256) void gemm_wmma_f32(const float* __restrict__ A,
                                                     const float* __restrict__ B,
                                                     float* __restrict__ C, int M) {
  constexpr int NT = N / 16;
  const int mTiles = (M + 15) >> 4;
  const int gwave = (int)((blockIdx.x * (long long)blockDim.x + threadIdx.x) >> 5);
  if (gwave >= mTiles * NT) return;  // wave-uniform: EXEC stays all-ones for WMMA

  const int lane = threadIdx.x & 31;
  const int mt = gwave / NT;
  const int nt = gwave - mt * NT;
  const int half = lane >> 4;  // selects K sub-pair (A) / M+8 rows (C)
  const int l16 = lane & 15;

  const int row = mt * 16 + l16;
  const int rclamp = row < M ? row : (M - 1);
  const int col = nt * 16 + l16;
  const float* __restrict__ Ar = A + (long long)rclamp * K;

  v8f acc = {};
#pragma unroll 4
  for (int k = 0; k < K; k += 4) {
    const int ka = k + 2 * half;
    v2f a, b;
    // A 16x4 f32 frag: lane l (l<16): M=l, VGPR0->K=ka, VGPR1->K=ka+1 (half=0)
    a.x = Ar[ka];
    a.y = Ar[ka + 1];
    // B 4x16 f32 frag: N = lane%16 striped across lanes, K per VGPR/half
    b.x = B[(long long)ka * N + col];
    b.y = B[(long long)(ka + 1) * N + col];
    acc = __builtin_amdgcn_wmma_f32_16x16x4_f32(false, a, false, b, (short)0, acc,
                                                false, false);
  }

  // C/D layout: VGPR v -> M = v + 8*(lane>=16), N = lane%16
  const int m0 = mt * 16 + 8 * half;           // first of 8 consecutive rows
  float* __restrict__ Cp = C + (long long)m0 * N + col;
  if (mt * 16 + 16 <= M) {
    // Full tile (always true when M % 16 == 0): unconditional coalesced stores.
#pragma unroll
    for (int v = 0; v < 8; ++v) Cp[(long long)v * N] = acc[v];
  } else {
    // Remainder tile (generic M): per-row guard.
#pragma unroll
    for (int v = 0; v < 8; ++v)
      if (m0 + v < M) Cp[(long long)v * N] = acc[v];
  }
}

// In-degree over real edges (self loop contributes +1 added in dinv kernel).
__global__ void deg_kernel(const int* __restrict__ dst, int* __restrict__ deg, int e) {
  const int i = blockIdx.x * blockDim.x + threadIdx.x;
  if (i < e) atomicAdd(&deg[dst[i]], 1);
}

__global__ void dinv_kernel(const int* __restrict__ deg, float* __restrict__ dinv, int n) {
  const int i = blockIdx.x * blockDim.x + threadIdx.x;
  if (i < n) dinv[i] = rsqrtf((float)deg[i] + 1.0f);  // deg>=1 incl. self loop
}

// out[dst, f] += H[src, f] * dinv[src]*dinv[dst];  edges >= e are self loops.
// One thread per (edge, feature); F = 1<<LOGF so no integer division.
template <int LOGF>
__global__ __launch_bounds__(256) void agg_kernel(const float* __restrict__ H,
                                                  const int* __restrict__ src,
                                                  const int* __restrict__ dst,
                                                  const float* __restrict__ dinv,
                                                  float* __restrict__ out, int e, int n) {
  constexpr int F = 1 << LOGF;
  const long long total = (long long)(e + n) << LOGF;
  const long long i = (long long)blockIdx.x * blockDim.x + threadIdx.x;
  if (i >= total) return;
  const int f = (int)(i & (F - 1));
  const long long edge = i >> LOGF;
  int s, d;
  if (edge < (long long)e) {
    s = src[edge];
    d = dst[edge];
  } else {
    s = d = (int)(edge - e);  // self loop
  }
  const float nrm = dinv[s] * dinv[d];
  atomicAdd(out + (((long long)d) << LOGF) + f, H[(((long long)s) << LOGF) + f] * nrm);
}

template <int LOGF>
__global__ void bias_relu_kernel(const float* __restrict__ agg, const float* __restrict__ b,
                                 float* __restrict__ out, long long total) {
  constexpr int F = 1 << LOGF;
  const long long i = (long long)blockIdx.x * blockDim.x + threadIdx.x;
  if (i >= total) return;
  const float v = agg[i] + b[(int)(i & (F - 1))];
  out[i] = v > 0.0f ? v : 0.0f;
}

// out[i,:] = log_softmax(agg[i,:] + b), 16 classes per node, 1 thread per node.
__global__ void lsm_kernel(const float* __restrict__ agg, const float* __restrict__ b,
                           float* __restrict__ out, int n) {
  const int i = blockIdx.x * blockDim.x + threadIdx.x;
  if (i >= n) return;
  float v[16];
  float mx = -INFINITY;
#pragma unroll
  for (int j = 0; j < 16; ++j) {
    v[j] = agg[(long long)i * 16 + j] + b[j];
    mx = fmaxf(mx, v[j]);
  }
  float s = 0.0f;
#pragma unroll
  for (int j = 0; j < 16; ++j) s += expf(v[j] - mx);
  const float lse = mx + logf(s);
#pragma unroll
  for (int j = 0; j < 16; ++j) out[(long long)i * 16 + j] = v[j] - lse;
}

extern "C" void kernel_launch(void* const* d_in, const int* in_sizes, int n_in,
                              void* d_out, int out_size, void* d_ws, size_t ws_size,
                              hipStream_t stream) {
  const float* x  = (const float*)d_in[0];  // [n, 256]
  const int*   ei = (const int*)d_in[1];    // [2, e]
  const float* W1 = (const float*)d_in[2];  // [256, 128]
  const float* b1 = (const float*)d_in[3];  // [128]
  const float* W2 = (const float*)d_in[4];  // [128, 16]
  const float* b2 = (const float*)d_in[5];  // [16]

  const int n = in_sizes[0] / 256;
  const int e = in_sizes[1] / 2;
  const int* src = ei;
  const int* dst = ei + e;

  // Workspace layout: [agg1 n*128 | agg2 n*16 | deg n]  <- zeroed together
  //                   [h1 n*128 | h2 n*16 | dinv n]     <- fully overwritten
  float* agg1 = (float*)d_ws;
  float* agg2 = agg1 + (size_t)n * 128;
  int*   deg  = (int*)(agg2 + (size_t)n * 16);
  float* h1   = (float*)(deg + n);
  float* h2   = h1 + (size_t)n * 128;
  float* dinv = h2 + (size_t)n * 16;

  // Zero accumulators + degree counters (one contiguous memset node).
  hipMemsetAsync(d_ws, 0, (size_t)n * (128 + 16 + 1) * sizeof(float), stream);

  // Degrees -> dinv.
  deg_kernel<<<ceil_div(e, 256), 256, 0, stream>>>(dst, deg, e);
  dinv_kernel<<<ceil_div(n, 256), 256, 0, stream>>>(deg, dinv, n);

  // Layer 1: GEMM (WMMA fp32), aggregate, bias+ReLU (reuse h1 as activated buf).
  {
    const long long waves = (long long)((n + 15) / 16) * (128 / 16);
    gemm_wmma_f32<256, 128><<<ceil_div(waves * 32, 256), 256, 0, stream>>>(x, W1, h1, n);
  }
  agg_kernel<7><<<ceil_div(((long long)(e + n)) << 7, 256), 256, 0, stream>>>(
      h1, src, dst, dinv, agg1, e, n);
  bias_relu_kernel<7><<<ceil_div((long long)n * 128, 256), 256, 0, stream>>>(
      agg1, b1, h1, (long long)n * 128);

  // Layer 2: GEMM (WMMA fp32), aggregate, log-softmax.
  {
    const long long waves = (long long)((n + 15) / 16) * (16 / 16);
    gemm_wmma_f32<128, 16><<<ceil_div(waves * 32, 256), 256, 0, stream>>>(h1, W2, h2, n);
  }
  agg_kernel<4><<<ceil_div(((long long)(e + n)) << 4, 256), 256, 0, stream>>>(
      h2, src, dst, dinv, agg2, e, n);
  lsm_kernel<<<ceil_div(n, 256), 256, 0, stream>>>(agg2, b2, (float*)d_out, n);
}